// BayerFeatureExtractor_74904229642986
// MI455X (gfx1250) — compile-verified
//
#include <hip/hip_runtime.h>
#include <cmath>
#include <cstring>

#ifndef M_PI
#define M_PI 3.14159265358979323846
#endif

typedef float v2f __attribute__((ext_vector_type(2)));
typedef float v8f __attribute__((ext_vector_type(8)));

#define TILE   16
#define HALO   4
#define IN_W   (TILE + 2*HALO)      // 24
#define R3_W   (TILE + 4)           // 20  (halo 2 region for f3 / fills)
#define NP3    (R3_W * R3_W)        // 400 = 25 * 16
#define NG3    (NP3 / 16)           // 25
#define NP5    (TILE * TILE)        // 256
#define NG5    (NP5 / 16)           // 16
#define IMG_W  768
#define EPSF   1e-6f

#if __has_builtin(__builtin_amdgcn_global_load_async_to_lds_b32) && \
    __has_builtin(__builtin_amdgcn_s_wait_asynccnt)
#define USE_ASYNC_LDS 1
#else
#define USE_ASYNC_LDS 0
#endif

struct Weights {
    float w3[16][12];   // [filter][tap], 10 filters x 9 taps, zero padded
    float w5[16][28];   // [filter][tap], 10 filters x 25 taps, zero padded
};

// ---------------------------------------------------------------------------
// Host-side weight construction (matches reference _build_kernels)
// ---------------------------------------------------------------------------
static void normk(float* k, int n) {
    double m = 0.0;
    for (int i = 0; i < n; ++i) m += k[i];
    m /= n;
    double s = 0.0;
    for (int i = 0; i < n; ++i) { k[i] = (float)(k[i] - m); s += fabs((double)k[i]); }
    if (s < 1e-6) s = 1e-6;
    for (int i = 0; i < n; ++i) k[i] = (float)(k[i] / s);
}

static void build_gabor(double theta, float* out) {
    const double sigma = 1.1, wl = 3.0, gamma = 0.65;
    for (int i = 0; i < 5; ++i) {
        for (int j = 0; j < 5; ++j) {
            double y = i - 2, x = j - 2;
            double xt =  x * cos(theta) + y * sin(theta);
            double yt = -x * sin(theta) + y * cos(theta);
            double env = exp(-(xt * xt + (gamma * yt) * (gamma * yt)) / (2.0 * sigma * sigma));
            out[i * 5 + j] = (float)(env * cos(2.0 * M_PI * xt / wl));
        }
    }
    normk(out, 25);
}

static void build_weights(Weights* W) {
    memset(W, 0, sizeof(*W));
    const float k3raw[10][9] = {
        {-1,0,1,-2,0,2,-1,0,1},                 // gx   (norm)
        {-1,-2,-1,0,0,0,1,2,1},                 // gy   (norm)
        {-2,-1,0,-1,0,1,0,1,2},                 // gdm  (norm)
        {0,1,2,-1,0,1,-2,-1,0},                 // gda  (norm)
        {0,1,0,1,-4,1,0,1,0},                   // lap  (raw)
        {0,0,0,1,-2,1,0,0,0},                   // hxx  (raw)
        {0,1,0,0,-2,0,0,1,0},                   // hyy  (raw)
        {0.25f,0,-0.25f,0,0,0,-0.25f,0,0.25f},  // hxy  (raw, /4)
        {0,0.25f,0,0.25f,0,0.25f,0,0.25f,0},    // gcross (raw)
        {1,-2,1,-2,4,-2,1,-2,1}                 // hf   (norm)
    };
    const bool dn3[10] = {true,true,true,true,false,false,false,false,false,true};
    for (int f = 0; f < 10; ++f) {
        float t[9];
        memcpy(t, k3raw[f], sizeof(t));
        if (dn3[f]) normk(t, 9);
        for (int i = 0; i < 9; ++i) W->w3[f][i] = t[i];
    }

    double s5[5], c5[5];
    for (int k = 0; k < 5; ++k) {
        s5[k] = sin(2.0 * M_PI * k / 5.0);
        c5[k] = cos(2.0 * M_PI * k / 5.0);
    }
    float t[25];
    // 0: checkerboard
    for (int i = 0; i < 5; ++i) for (int j = 0; j < 5; ++j) t[i*5+j] = ((i + j) & 1) ? -1.f : 1.f;
    normk(t, 25); for (int i = 0; i < 25; ++i) W->w5[0][i] = t[i];
    // 1: sh (column parity)
    for (int i = 0; i < 5; ++i) for (int j = 0; j < 5; ++j) t[i*5+j] = (j & 1) ? -1.f : 1.f;
    normk(t, 25); for (int i = 0; i < 25; ++i) W->w5[1][i] = t[i];
    // 2: sv (row parity)
    for (int i = 0; i < 5; ++i) for (int j = 0; j < 5; ++j) t[i*5+j] = (i & 1) ? -1.f : 1.f;
    normk(t, 25); for (int i = 0; i < 25; ++i) W->w5[2][i] = t[i];
    // 3: sx = s along columns
    for (int i = 0; i < 5; ++i) for (int j = 0; j < 5; ++j) t[i*5+j] = (float)s5[j];
    normk(t, 25); for (int i = 0; i < 25; ++i) W->w5[3][i] = t[i];
    // 4: sy = s along rows
    for (int i = 0; i < 5; ++i) for (int j = 0; j < 5; ++j) t[i*5+j] = (float)s5[i];
    normk(t, 25); for (int i = 0; i < 25; ++i) W->w5[4][i] = t[i];
    // 5: px = c along columns
    for (int i = 0; i < 5; ++i) for (int j = 0; j < 5; ++j) t[i*5+j] = (float)c5[j];
    normk(t, 25); for (int i = 0; i < 25; ++i) W->w5[5][i] = t[i];
    // 6: py = c along rows
    for (int i = 0; i < 5; ++i) for (int j = 0; j < 5; ++j) t[i*5+j] = (float)c5[i];
    normk(t, 25); for (int i = 0; i < 25; ++i) W->w5[6][i] = t[i];
    // 7: gabor 45, 8: gabor 135
    build_gabor(M_PI / 4.0, t);        for (int i = 0; i < 25; ++i) W->w5[7][i] = t[i];
    build_gabor(3.0 * M_PI / 4.0, t);  for (int i = 0; i < 25; ++i) W->w5[8][i] = t[i];
    // 9: DCT(u=2,v=2)
    for (int i = 0; i < 5; ++i)
        for (int j = 0; j < 5; ++j)
            t[i*5+j] = (float)(cos(M_PI * (i + 0.5) * 2.0 / 5.0) * cos(M_PI * (j + 0.5) * 2.0 / 5.0));
    normk(t, 25); for (int i = 0; i < 25; ++i) W->w5[9][i] = t[i];
}

// ---------------------------------------------------------------------------
// Device kernel
// ---------------------------------------------------------------------------
__device__ __forceinline__ int refl(int v, int n) {
    if (v < 0) v = -v;
    if (v >= n) v = 2 * n - 2 - v;
    return v;
}

__global__ __launch_bounds__(256)
void bayer_feat_kernel(const float* __restrict__ in, float* __restrict__ out,
                       const Weights* __restrict__ W) {
    __shared__ float s_in[IN_W * IN_W];       // input tile, coords -4..19
    __shared__ float s_f3[NP3 * 10];          // f3 region (-2..17), 10 ch
    __shared__ float s_f5[NP5 * 10];          // f5 center (0..15), 10 ch
    __shared__ float s_cd[NP3 * 2];           // rg,bg on region (-2..17)
    __shared__ float s_gpd[TILE * TILE];      // fgr - fgb on center

    const int tx0 = blockIdx.x * TILE;
    const int ty0 = blockIdx.y * TILE;
    const int b   = blockIdx.z;
    const int tid = threadIdx.x;
    const int lane = tid & 31;
    const int wave = tid >> 5;
    const size_t plane = (size_t)IMG_W * IMG_W;
    const float* img = in + (size_t)b * plane;

    // ---- Stage 1: stage input tile (reflect padding) ----------------------
    for (int p = tid; p < IN_W * IN_W; p += 256) {
        int ly = p / IN_W, lx = p % IN_W;
        int gy = refl(ty0 + ly - HALO, IMG_W);
        int gx = refl(tx0 + lx - HALO, IMG_W);
        const float* src = &img[(size_t)gy * IMG_W + gx];
#if USE_ASYNC_LDS
        __builtin_amdgcn_global_load_async_to_lds_b32(
            (__attribute__((address_space(1))) int*)const_cast<float*>(src),
            (__attribute__((address_space(3))) int*)&s_in[p], 0, 0);
#else
        s_in[p] = *src;
#endif
    }
#if USE_ASYNC_LDS
    __builtin_amdgcn_s_wait_asynccnt(0);
#endif
    __syncthreads();

    // ---- Stage 2: multi-filter convs on the matrix units ------------------
    // V_WMMA_F32_16X16X4_F32: A = 16 pixels x 4 taps, B = 4 taps x 16 filters.
    // Out-of-range K taps carry ZERO weights in w3/w5, so A-addresses are just
    // clamped to a valid tap (finite * 0 == 0): all LDS loads unconditional,
    // no EXEC divergence around the WMMA chain.
    const int n     = lane & 15;            // filter column
    const int khalf = (lane >> 4) << 1;     // K sub-pair: 0 (lanes 0-15), 2 (lanes 16-31)
    const int mrow  = lane & 15;            // pixel row of A held by this lane

    v2f b3[3], b5[7];
#pragma unroll
    for (int c = 0; c < 3; ++c) {
        b3[c].x = W->w3[n][4 * c + khalf];
        b3[c].y = W->w3[n][4 * c + khalf + 1];
    }
#pragma unroll
    for (int c = 0; c < 7; ++c) {
        b5[c].x = W->w5[n][4 * c + khalf];
        b5[c].y = W->w5[n][4 * c + khalf + 1];
    }

    // f3 over 20x20 region (25 groups of 16 pixels)
    for (int g = wave; g < NG3; g += 8) {
        int p    = g * 16 + mrow;
        int ry   = p / R3_W;
        int rx   = p % R3_W;
        int base = (ry + 2) * IN_W + (rx + 2);   // region origin (-2,-2) in tile
        v2f a3[3];
#pragma unroll
        for (int c = 0; c < 3; ++c) {
            int t0 = 4 * c + khalf;     if (t0 > 8) t0 = 8;
            int t1 = 4 * c + khalf + 1; if (t1 > 8) t1 = 8;
            a3[c].x = s_in[base + (t0 / 3 - 1) * IN_W + (t0 % 3 - 1)];
            a3[c].y = s_in[base + (t1 / 3 - 1) * IN_W + (t1 % 3 - 1)];
        }
        v8f acc = {};
#pragma unroll
        for (int c = 0; c < 3; ++c)
            acc = __builtin_amdgcn_wmma_f32_16x16x4_f32(false, a3[c], false, b3[c],
                                                        (short)0, acc, false, false);
        if (n < 10) {
            int pmb = g * 16 + ((lane >> 4) << 3);
#pragma unroll
            for (int r = 0; r < 8; ++r)
                s_f3[(pmb + r) * 10 + n] = acc[r];
        }
    }

    // f5 over 16x16 center (16 groups of 16 pixels)
    for (int g = wave; g < NG5; g += 8) {
        int p    = g * 16 + mrow;
        int ry   = p / TILE;
        int rx   = p % TILE;
        int base = (ry + HALO) * IN_W + (rx + HALO);
        v2f a5[7];
#pragma unroll
        for (int c = 0; c < 7; ++c) {
            int t0 = 4 * c + khalf;     if (t0 > 24) t0 = 24;
            int t1 = 4 * c + khalf + 1; if (t1 > 24) t1 = 24;
            a5[c].x = s_in[base + (t0 / 5 - 2) * IN_W + (t0 % 5 - 2)];
            a5[c].y = s_in[base + (t1 / 5 - 2) * IN_W + (t1 % 5 - 2)];
        }
        v8f acc = {};
#pragma unroll
        for (int c = 0; c < 7; ++c)
            acc = __builtin_amdgcn_wmma_f32_16x16x4_f32(false, a5[c], false, b5[c],
                                                        (short)0, acc, false, false);
        if (n < 10) {
            int pmb = g * 16 + ((lane >> 4) << 3);
#pragma unroll
            for (int r = 0; r < 8; ++r)
                s_f5[(pmb + r) * 10 + n] = acc[r];
        }
    }
    __syncthreads();

    // ---- Stage 3: Bayer fills (SMOOTH5; denominators analytic) ------------
    // SMOOTH5 = outer([1,2,3,2,1],[1,2,3,2,1])/81. Mask conv denominators are
    // exact 2x2-parity lookups (reflect padding preserves parity).
    {
        const float swv[5] = {1.f, 2.f, 3.f, 2.f, 1.f};
        for (int p = tid; p < NP3; p += 256) {
            int ry = p / R3_W - 2, rx = p % R3_W - 2;
            int gy = ty0 + ry, gx = tx0 + rx;
            int base = (ry + HALO) * IN_W + (rx + HALO);
            float nr = 0.f, nb = 0.f, ngr = 0.f, ngb = 0.f;
#pragma unroll
            for (int dy = -2; dy <= 2; ++dy) {
#pragma unroll
                for (int dx = -2; dx <= 2; ++dx) {
                    float w = swv[dy + 2] * swv[dx + 2] * (1.f / 81.f);
                    float v = s_in[base + dy * IN_W + dx] * w;
                    int pr = (gy + dy) & 1, pc = (gx + dx) & 1;
                    if (pr) { if (pc) ngr += v; else nr += v; }
                    else    { if (pc) nb  += v; else ngb += v; }
                }
            }
            float ei = ((gy & 1) == 0) ? (5.f / 9.f) : (4.f / 9.f);
            float ej = ((gx & 1) == 0) ? (5.f / 9.f) : (4.f / 9.f);
            float oi = 1.f - ei, oj = 1.f - ej;
            float den_r  = oi * ej, den_b = ei * oj;
            float den_gr = oi * oj, den_gb = ei * ej;
            float fr = nr / fmaxf(den_r, EPSF);
            float fb = nb / fmaxf(den_b, EPSF);
            float fg = (ngr + ngb) / fmaxf(den_gr + den_gb, EPSF);
            s_cd[p * 2 + 0] = fr - fg;
            s_cd[p * 2 + 1] = fb - fg;
            if ((unsigned)ry < TILE && (unsigned)rx < TILE) {
                float fgr = ngr / fmaxf(den_gr, EPSF);
                float fgb = ngb / fmaxf(den_gb, EPSF);
                s_gpd[ry * TILE + rx] = fgr - fgb;
            }
        }
    }
    __syncthreads();

    // ---- Stage 4: per-pixel feature assembly ------------------------------
    {
        const float swv[5] = {1.f, 2.f, 3.f, 2.f, 1.f};
        const int y = tid / TILE, x = tid % TILE;
        const int gy = ty0 + y, gx = tx0 + x;
        const int r3 = (y + 2) * R3_W + (x + 2);
        const float* F3 = &s_f3[r3 * 10];
        const float gxv = F3[0], gyv = F3[1], gdm = F3[2], gda = F3[3];
        const float lap = F3[4], hxx = F3[5], hyy = F3[6], hxy = F3[7];
        const float gcross = F3[8], hf = F3[9];
        const float* F5 = &s_f5[(y * TILE + x) * 10];
        const float cbv = F5[0], shv = F5[1], svv = F5[2];
        const float sxv = F5[3], syv = F5[4], pxv = F5[5], pyv = F5[6];
        const float g45 = F5[7], g135 = F5[8], dct = F5[9];

        // structure tensor J (SMOOTH5 over gx^2, gy^2, gx*gy)
        float Jxx = 0.f, Jyy = 0.f, Jxy = 0.f;
#pragma unroll
        for (int dy = -2; dy <= 2; ++dy) {
#pragma unroll
            for (int dx = -2; dx <= 2; ++dx) {
                int q = ((y + 2 + dy) * R3_W + (x + 2 + dx)) * 10;
                float a = s_f3[q], bb = s_f3[q + 1];
                float w = swv[dy + 2] * swv[dx + 2] * (1.f / 81.f);
                Jxx += w * a * a; Jyy += w * bb * bb; Jxy += w * a * bb;
            }
        }
        // BOX5 chroma variance
        float m1r = 0.f, m1b = 0.f, m2r = 0.f, m2b = 0.f;
#pragma unroll
        for (int dy = -2; dy <= 2; ++dy) {
#pragma unroll
            for (int dx = -2; dx <= 2; ++dx) {
                int q = ((y + 2 + dy) * R3_W + (x + 2 + dx)) * 2;
                float a = s_cd[q], bb = s_cd[q + 1];
                m1r += a; m1b += bb; m2r += a * a; m2b += bb * bb;
            }
        }
        m1r *= 0.04f; m1b *= 0.04f; m2r *= 0.04f; m2b *= 0.04f;
        float cdv = fmaxf(m2r - m1r * m1r, 0.f) + fmaxf(m2b - m1b * m1b, 0.f);

        // Haar gradients
        const float ha[5] = {-0.25f, 0.5f, 0.5f, 0.5f, -0.25f};
        const int bi = (y + HALO) * IN_W + (x + HALO);
        float gh = 0.f, gv = 0.f;
#pragma unroll
        for (int k = 0; k < 5; ++k) {
            gh += ha[k] * s_in[bi + (k - 2)];
            gv += ha[k] * s_in[bi + (k - 2) * IN_W];
        }
        float dgc = fabsf(gh - gv);

        // pointwise features
        float grad_mag = sqrtf(gxv * gxv + gyv * gyv + EPSF);
        float hm = 0.5f * (hxx + hyy);
        float hs = sqrtf(0.25f * (hxx - hyy) * (hxx - hyy) + hxy * hxy + EPSF);
        float lam_max = hm + hs, lam_min = hm - hs;
        float aniso = 2.f * sqrtf(0.25f * (Jxx - Jyy) * (Jxx - Jyy) + Jxy * Jxy + EPSF)
                      / (Jxx + Jyy + EPSF);
        float dmean = 0.25f * (gxv + gyv + gdm + gda);
        float dsq   = 0.25f * (gxv * gxv + gyv * gyv + gdm * gdm + gda * gda);
        float dir_var = dsq - dmean * dmean;
        float orient_e = sqrtf(g45 * g45 + g135 * g135 + EPSF);
        float phase_e  = sqrtf(pxv * pxv + pyv * pyv + EPSF);
        float highband = sqrtf(dct * dct + hf * hf + EPSF);

        int er = ((gy & 1) == 0), ec = ((gx & 1) == 0);
        float r_m = (float)(!er &&  ec);
        float b_m = (float)( er && !ec);
        float g_m = (float)( er ==  ec);

        float rg  = s_cd[r3 * 2 + 0];
        float bg  = s_cd[r3 * 2 + 1];
        float gpd = s_gpd[y * TILE + x];
        float chroma = sqrtf(rg * rg + bg * bg + EPSF);
        float gir = (s_in[bi] - gcross) * g_m;

        float feats[30];
        feats[0]  = gxv;      feats[1]  = gyv;     feats[2]  = gdm;     feats[3]  = gda;
        feats[4]  = grad_mag; feats[5]  = lap;     feats[6]  = lam_max; feats[7]  = lam_min;
        feats[8]  = aniso;    feats[9]  = dir_var; feats[10] = orient_e;
        feats[11] = r_m;      feats[12] = g_m;     feats[13] = b_m;
        feats[14] = gpd;      feats[15] = rg;      feats[16] = bg;      feats[17] = gir;
        feats[18] = dgc;      feats[19] = chroma;  feats[20] = cdv;
        feats[21] = fabsf(cbv); feats[22] = shv;   feats[23] = svv;
        feats[24] = shv;      feats[25] = svv;     feats[26] = phase_e;
        feats[27] = sxv;      feats[28] = syv;     feats[29] = highband;

        size_t o0 = (size_t)b * 30 * plane + (size_t)gy * IMG_W + gx;
#pragma unroll
        for (int f = 0; f < 30; ++f)
            out[o0 + (size_t)f * plane] = feats[f];
    }
}

// ---------------------------------------------------------------------------
// Launch
// ---------------------------------------------------------------------------
extern "C" void kernel_launch(void* const* d_in, const int* in_sizes, int n_in,
                              void* d_out, int out_size, void* d_ws, size_t ws_size,
                              hipStream_t stream) {
    (void)in_sizes; (void)n_in; (void)out_size; (void)ws_size;
    static Weights h_w;
    build_weights(&h_w);
    (void)hipMemcpyAsync(d_ws, &h_w, sizeof(Weights), hipMemcpyHostToDevice, stream);

    const float* bayer = (const float*)d_in[0];
    float* out = (float*)d_out;
    dim3 grid(IMG_W / TILE, IMG_W / TILE, 4);
    bayer_feat_kernel<<<grid, 256, 0, stream>>>(bayer, out, (const Weights*)d_ws);
}